// MRConv2d_60997125538360
// MI455X (gfx1250) — compile-verified
//
#include <hip/hip_runtime.h>

typedef __attribute__((ext_vector_type(2))) float v2f;
typedef __attribute__((ext_vector_type(8))) float v8f;

#define TILE_N    128
#define LDS_STRIDE 144   // 128 + 16 pad: B-frag rows r and r+2 hit disjoint bank groups
#define C_INx     64
#define C2x       128
#define C_OUTx    128
#define NNODES    16384
#define KNBR      9

__global__ __launch_bounds__(256)
void mrconv_wmma_fused(const float* __restrict__ x,
                       const int*   __restrict__ eidx,
                       const float* __restrict__ W,
                       const float* __restrict__ bias,
                       float*       __restrict__ out)
{
    __shared__ float feat[C2x * LDS_STRIDE];   // 72 KB: [channel][node] tile

    const int tid = threadIdx.x;
    const int bid = blockIdx.x;
    const int b   = bid >> 7;                  // bid / (N / TILE_N)
    const int n0  = (bid & 127) * TILE_N;

    const float* xb = x + (size_t)b * C_INx * NNODES;

    // ---- Stage 1a: coalesced copy of x (channels 0..63) into LDS ----
    for (int i = tid; i < C_INx * (TILE_N / 4); i += 256) {
        const int c  = i >> 5;                 // / (TILE_N/4)
        const int v4 = (i & 31) << 2;
        float4 v = *(const float4*)(xb + (size_t)c * NNODES + n0 + v4);
        *(float4*)(&feat[c * LDS_STRIDE + v4]) = v;
    }
    __syncthreads();

    // ---- Stage 1b: max-relative aggregation -> LDS rows 64..127 ----
    {
        const int node = tid >> 1;
        const int half = tid & 1;
        int nbr[KNBR];
        const int* ep = eidx + ((size_t)b * NNODES + (n0 + node)) * KNBR;
        #pragma unroll
        for (int k = 0; k < KNBR; ++k) nbr[k] = ep[k];

        for (int c = 0; c < 32; ++c) {
            const int cc = half * 32 + c;
            const float* xc = xb + (size_t)cc * NNODES;
            const float xi = feat[cc * LDS_STRIDE + node];
            float m = -3.402823466e38f;
            #pragma unroll
            for (int k = 0; k < KNBR; ++k)
                m = fmaxf(m, xc[nbr[k]] - xi);   // random 4B reads, L2-resident
            feat[(C_INx + cc) * LDS_STRIDE + node] = m;
        }
    }
    __syncthreads();

    // ---- Stage 2: out[128 x 128] = W[128x128] * feat[128x128] + b, ReLU ----
    const int wave = tid >> 5;                 // 0..7: 16-row output strip
    const int lane = tid & 31;
    const int hi   = lane >> 4;                // 0: K even pair low, 1: K pair +2
    const int col  = lane & 15;
    const int mrow = wave * 16 + col;          // A-matrix row owned by this lane

    // Preload full 16x128 strip of W as 32 A-fragments (16x4 f32 layout):
    // VGPR0 = K = 4*kk + 2*hi, VGPR1 = K+1
    v2f a[32];
    #pragma unroll
    for (int kk = 0; kk < 32; ++kk) {
        const float* wp = W + (size_t)mrow * C2x + kk * 4 + hi * 2;
        a[kk].x = wp[0];
        a[kk].y = wp[1];
    }
    // Bias per accumulator VGPR i -> output row wave*16 + 8*hi + i
    float bv[8];
    #pragma unroll
    for (int i = 0; i < 8; ++i) bv[i] = bias[wave * 16 + hi * 8 + i];

    for (int nt = 0; nt < TILE_N; nt += 16) {
        v8f acc = {};
        #pragma unroll
        for (int kk = 0; kk < 32; ++kk) {
            // B 4x16 f32 frag: VGPR0 row K+2*hi, VGPR1 row K+1+2*hi, col = lane&15
            const int r = kk * 4 + hi * 2;
            v2f bf;
            bf.x = feat[(r    ) * LDS_STRIDE + nt + col];
            bf.y = feat[(r + 1) * LDS_STRIDE + nt + col];
            acc = __builtin_amdgcn_wmma_f32_16x16x4_f32(
                false, a[kk], false, bf, (short)0, acc, false, false);
        }
        // C/D layout: VGPR i -> row (wave*16 + 8*hi + i), node nt + col
        float* op = out + ((size_t)b * C_OUTx + wave * 16 + hi * 8) * NNODES
                        + n0 + nt + col;
        #pragma unroll
        for (int i = 0; i < 8; ++i)
            op[(size_t)i * NNODES] = fmaxf(acc[i] + bv[i], 0.0f);
    }
}

extern "C" void kernel_launch(void* const* d_in, const int* in_sizes, int n_in,
                              void* d_out, int out_size, void* d_ws, size_t ws_size,
                              hipStream_t stream) {
    const float* x    = (const float*)d_in[0];   // [8, 64, 16384, 1]
    const int*   eidx = (const int*)  d_in[1];   // [2, 8, 16384, 9] (plane 0 used)
    const float* W    = (const float*)d_in[2];   // [128, 128]
    const float* bias = (const float*)d_in[3];   // [128]
    float*       out  = (float*)d_out;           // [8, 128, 16384, 1]

    const int blocks = 8 * (NNODES / TILE_N);    // 1024
    mrconv_wmma_fused<<<blocks, 256, 0, stream>>>(x, eidx, W, bias, out);
}